// CausalNet_635655160379
// MI455X (gfx1250) — compile-verified
//
#include <hip/hip_runtime.h>
#include <hip/hip_bf16.h>
#include <stdint.h>

// Problem constants (fixed by the reference harness).
#define T_LEN 1024
#define B_OPT 32
#define PIPE  4  // async double-buffer depth (ring of 4 LDS buffers)

// ---------------------------------------------------------------------------
// CDNA5 async global->LDS copy (16 bytes per lane) + async counter waits.
// the LDS address comes from the vdst VGPR; generic shared-pointer low 32 bits are the
// LDS byte offset, so truncating the flat address gives the VDST value.
// ---------------------------------------------------------------------------
__device__ __forceinline__ void async_cp16(unsigned lds_byte_off, const float* gsrc) {
  asm volatile("global_load_async_to_lds_b128 %0, %1, off"
               :
               : "v"(lds_byte_off), "v"(gsrc)
               : "memory");
}
__device__ __forceinline__ void wait_async_3() {
  asm volatile("s_wait_asynccnt 0x3" ::: "memory");
}
__device__ __forceinline__ void wait_async_n(int n) {
  if (n <= 0)      asm volatile("s_wait_asynccnt 0x0" ::: "memory");
  else if (n == 1) asm volatile("s_wait_asynccnt 0x1" ::: "memory");
  else if (n == 2) asm volatile("s_wait_asynccnt 0x2" ::: "memory");
  else             asm volatile("s_wait_asynccnt 0x3" ::: "memory");
}

// log(exp(a)+exp(b)), safe when one side is -inf (max stays finite).
__device__ __forceinline__ float lae(float a, float b) {
  float m = fmaxf(a, b);
  if (m == -INFINITY) return -INFINITY;
  return m + __logf(__expf(a - m) + __expf(b - m));
}

// logsumexp across the 32 lanes of a wave (result broadcast to all lanes).
__device__ __forceinline__ float lse_wave(float x) {
  float m = x;
#pragma unroll
  for (int d = 16; d >= 1; d >>= 1) m = fmaxf(m, __shfl_xor(m, d, 32));
  float e = __expf(x - m);
#pragma unroll
  for (int d = 16; d >= 1; d >>= 1) e += __shfl_xor(e, d, 32);
  return m + __logf(e);
}

// ---------------------------------------------------------------------------
// Kernel 1: sequential forward + backward scans, O(T*b). One wave; lane = opt.
// The reference's O(T^2*b) forward DP factorizes because only channel 0
// propagates and all states share the same per-step increment:
//   fw_{t}[o,s,1] = D_s[o] + A_t[o] + delta_t[o]   (s <= t-1)
//   D[s][o]  = LSE1_s + start_s[o] - A_s[o]
//   V[i][o]  = A_{i+1}[o] + delta_{i+1}[o] + c_{i+1} - total_logp
//   tlp_out  = total_logp = LSE1_T
// stop is stop_logps with last dim reversed: stop0 = sl[..,1], stop1 = sl[..,0].
// ---------------------------------------------------------------------------
__global__ __launch_bounds__(32) void cc_scan_kernel(
    const float* __restrict__ act,   // (T, B)
    const float* __restrict__ sl,    // (T+1, B, 2)
    const float* __restrict__ st,    // (T+1, B)
    float* __restrict__ D,           // (T, B)
    float* __restrict__ V,           // (T, B)
    float* __restrict__ tlp_out) {
  const int o = threadIdx.x;  // 0..31
  __shared__ float c_arr[T_LEN];  // c_arr[i] = c_{i+1}

  // ---- forward scan ----
  float A = 0.0f;
  float M = -INFINITY;
  float Dprev = st[o];  // D_0 = start_0 (LSE1_0 = 0, A_0 = 0)
  D[o] = Dprev;
  float lse1 = 0.0f;
  for (int t = 1; t <= T_LEN; ++t) {
    float a = act[(t - 1) * B_OPT + o];
    float2 s2 = *reinterpret_cast<const float2*>(sl + ((size_t)t * B_OPT + o) * 2);
    float stop0 = s2.y;            // reversed channel 0
    float delta = s2.x - s2.y;     // stop1 - stop0
    A += a + stop0;
    M = lae(M, Dprev);             // M_t = lse_{s<t} D_s
    lse1 = lse_wave(A + delta + M);
    if (t < T_LEN) {
      float Dt = lse1 + st[t * B_OPT + o] - A;
      D[t * B_OPT + o] = Dt;
      Dprev = Dt;
    }
  }
  const float tlp = lse1;  // total_logp
  if (o == 0) tlp_out[0] = tlp;

  // ---- backward scan ----
  float h = -INFINITY;  // channel-0 value
  float cn = 0.0f;      // c[T]
  if (o == 0) c_arr[T_LEN - 1] = 0.0f;
  for (int t = T_LEN - 1; t >= 0; --t) {
    float2 s2 = *reinterpret_cast<const float2*>(sl + ((size_t)(t + 1) * B_OPT + o) * 2);
    float q = lae(s2.y + h, s2.x + cn);  // stop0+h  vs  stop1+c
    h = act[t * B_OPT + o] + q;
    float ct = lse_wave(st[t * B_OPT + o] + h);  // scalar, broadcast
    if (o == 0 && t >= 1) c_arr[t - 1] = ct;
    cn = ct;
  }
  __syncthreads();

  // ---- emit V (recompute prefix A; avoids 128 KB extra scratch) ----
  float A2 = 0.0f;
  for (int t = 1; t <= T_LEN; ++t) {
    float a = act[(t - 1) * B_OPT + o];
    float2 s2 = *reinterpret_cast<const float2*>(sl + ((size_t)t * B_OPT + o) * 2);
    A2 += a + s2.y;
    float delta = s2.x - s2.y;
    V[(t - 1) * B_OPT + o] = A2 + delta + c_arr[t - 1] - tlp;
  }
}

// ---------------------------------------------------------------------------
// Kernel 2: total_cc partial per segment-start s (one block per s).
// Streams causal_pens[s, i+1, 0:32] rows (contiguous 128B each) for i=s..T-1
// through LDS with a 4-deep ring of async-to-LDS copies. Each thread owns 4
// options (o4..o4+3) of one row per 32-row chunk and consumes exactly the
// 16B it loaded -> no barriers needed, only per-wave s_wait_asynccnt.
// exp(D_s+V_i) is fused per element: the combined exponent is log(w) <= 0,
// so it cannot overflow (the separated factors can).
// ---------------------------------------------------------------------------
__global__ __launch_bounds__(256) void cc_reduce_kernel(
    const float* __restrict__ cp,  // (T+1, T+1, B)
    const float* __restrict__ D,   // (T, B)
    const float* __restrict__ V,   // (T, B)  (128 KB -> stays L2-resident)
    float* __restrict__ partials)  // (T)
{
  const int s = blockIdx.x;
  const int tid = threadIdx.x;
  const int o4 = (tid & 7) * 4;  // option group
  const int r = tid >> 3;        // row within 32-row chunk

  __shared__ __align__(16) float lbuf[PIPE * 256 * 4];  // 16 KB ring
  __shared__ float wsum[8];

  const unsigned lb = (unsigned)(uintptr_t)(&lbuf[0]);  // LDS byte offset
  // Row i of this block's slab starts at cp[(s*(T+1) + i + 1)*B + o4].
  const float* cpb = cp + ((size_t)s * (T_LEN + 1) + 1) * B_OPT + o4;
  const float* vb  = V + o4;
  const float4 d4 = *reinterpret_cast<const float4*>(D + (size_t)s * B_OPT + o4);

  const int nrows = T_LEN - s;
  const int nchunk = (nrows + 31) >> 5;

  float4 acc = make_float4(0.f, 0.f, 0.f, 0.f);

  // Prime the ring (rows past T-1 stay inside the (T+1)^2*B buffer, so
  // over-fetch is safe; they are masked out in the accumulate step).
  for (int k = 0; k < PIPE && k < nchunk; ++k) {
    int i = s + k * 32 + r;
    async_cp16(lb + (unsigned)((k & (PIPE - 1)) * 4096) + (unsigned)tid * 16u,
               cpb + (size_t)i * B_OPT);
  }

  int k = 0;
  // Steady state: 4 chunks in flight, branch-free wait.
  for (; k + PIPE < nchunk; ++k) {
    wait_async_3();  // chunk k has landed
    const int i = s + k * 32 + r;
    {
      const float4 c4 = *reinterpret_cast<const float4*>(
          &lbuf[(size_t)(k & (PIPE - 1)) * 1024 + (size_t)tid * 4]);
      const float4 v4 = *reinterpret_cast<const float4*>(vb + (size_t)i * B_OPT);
      acc.x += __expf(d4.x + v4.x) * c4.x;
      acc.y += __expf(d4.y + v4.y) * c4.y;
      acc.z += __expf(d4.z + v4.z) * c4.z;
      acc.w += __expf(d4.w + v4.w) * c4.w;
    }
    const int inext = i + PIPE * 32;
    // Pull the upcoming V row toward the WGP (global_prefetch_b8).
    __builtin_prefetch(vb + (size_t)inext * B_OPT, 0, 3);
    async_cp16(lb + (unsigned)((k & (PIPE - 1)) * 4096) + (unsigned)tid * 16u,
               cpb + (size_t)inext * B_OPT);
  }
  // Drain.
  for (; k < nchunk; ++k) {
    wait_async_n(nchunk - k - 1);
    const int i = s + k * 32 + r;
    if (i < T_LEN) {
      const float4 c4 = *reinterpret_cast<const float4*>(
          &lbuf[(size_t)(k & (PIPE - 1)) * 1024 + (size_t)tid * 4]);
      const float4 v4 = *reinterpret_cast<const float4*>(vb + (size_t)i * B_OPT);
      acc.x += __expf(d4.x + v4.x) * c4.x;
      acc.y += __expf(d4.y + v4.y) * c4.y;
      acc.z += __expf(d4.z + v4.z) * c4.z;
      acc.w += __expf(d4.w + v4.w) * c4.w;
    }
  }

  // Deterministic fixed-order block reduction.
  float sum = (acc.x + acc.y) + (acc.z + acc.w);
#pragma unroll
  for (int d = 16; d >= 1; d >>= 1) sum += __shfl_xor(sum, d, 32);
  if ((tid & 31) == 0) wsum[tid >> 5] = sum;
  __syncthreads();
  if (tid == 0) {
    float t = 0.f;
#pragma unroll
    for (int w = 0; w < 8; ++w) t += wsum[w];
    partials[s] = t;
  }
}

// ---------------------------------------------------------------------------
// Kernel 3: loss = -total_logp + CC_WEIGHT * total_cc (fixed-order sum).
// ---------------------------------------------------------------------------
__global__ __launch_bounds__(256) void cc_final_kernel(
    const float* __restrict__ partials, const float* __restrict__ tlp,
    float* __restrict__ out) {
  __shared__ float red[256];
  float s = 0.f;
  for (int j = threadIdx.x; j < T_LEN; j += 256) s += partials[j];
  red[threadIdx.x] = s;
  __syncthreads();
  for (int step = 128; step >= 1; step >>= 1) {
    if (threadIdx.x < step) red[threadIdx.x] += red[threadIdx.x + step];
    __syncthreads();
  }
  if (threadIdx.x == 0) out[0] = -tlp[0] + red[0];
}

// ---------------------------------------------------------------------------
extern "C" void kernel_launch(void* const* d_in, const int* in_sizes, int n_in,
                              void* d_out, int out_size, void* d_ws,
                              size_t ws_size, hipStream_t stream) {
  (void)in_sizes; (void)n_in; (void)out_size; (void)ws_size;
  const float* act = (const float*)d_in[0];  // (1024, 32)
  const float* sl  = (const float*)d_in[1];  // (1025, 32, 2)
  const float* st  = (const float*)d_in[2];  // (1025, 32)
  const float* cp  = (const float*)d_in[3];  // (1025, 1025, 32)
  float* out = (float*)d_out;

  // Scratch layout (floats): D[32768] | V[32768] | tlp[1] pad | partials[1024]
  float* ws = (float*)d_ws;
  float* D = ws;
  float* V = ws + (size_t)T_LEN * B_OPT;
  float* tlp = ws + (size_t)2 * T_LEN * B_OPT;
  float* partials = ws + (size_t)2 * T_LEN * B_OPT + 4;

  cc_scan_kernel<<<1, 32, 0, stream>>>(act, sl, st, D, V, tlp);
  cc_reduce_kernel<<<T_LEN, 256, 0, stream>>>(cp, D, V, partials);
  cc_final_kernel<<<1, 256, 0, stream>>>(partials, tlp, out);
}